// GATEAU_12558484373813
// MI455X (gfx1250) — compile-verified
//
#include <hip/hip_runtime.h>
#include <hip/hip_bf16.h>

// ---------------------------------------------------------------------------
// GAT-style message passing for MI455X (gfx1250, wave32, WMMA).
// Memory-bound (AI ~16 F/B): use exact fp32 WMMA (V_WMMA_F32_16X16X4_F32),
// single fused pass over the 205MB edge array.
// ---------------------------------------------------------------------------

#define N_NODES 50000
#define N_EDGES 800000
#define DIM     64

typedef __attribute__((ext_vector_type(2))) float v2f;
typedef __attribute__((ext_vector_type(8))) float v8f;

// Monotone order-preserving encoding of float into uint32 for atomicMax.
__device__ __forceinline__ unsigned enc_f32(float f) {
    unsigned u = __float_as_uint(f);
    return (u & 0x80000000u) ? ~u : (u | 0x80000000u);
}
__device__ __forceinline__ float dec_f32(unsigned k) {
    unsigned u = (k & 0x80000000u) ? (k & 0x7FFFFFFFu) : ~k;
    return __uint_as_float(u);
}

// --------------------------------------------------------------------------
// Kernel 0: init segment-max keys (encoded -inf == 0) and segment sums.
// --------------------------------------------------------------------------
__global__ void gat_init_kernel(unsigned* __restrict__ maxkeys,
                                float* __restrict__ sums) {
    int i = blockIdx.x * blockDim.x + threadIdx.x;
    if (i < N_NODES) {
        maxkeys[i] = 0u;
        sums[i]    = 0.0f;
    }
}

// --------------------------------------------------------------------------
// Kernel 1: node projections. 4 waves/block; wave w multiplies a 16x64 node
// tile by one of the 4 weight matrices using 64x v_wmma_f32_16x16x4_f32.
// Wave 3 (self projection) writes directly into d_out as the accumulator
// seed for new_nodes.
// --------------------------------------------------------------------------
__global__ void __launch_bounds__(128)
gat_node_proj_kernel(const float* __restrict__ nodes,
                     const float* __restrict__ Wsent, const float* __restrict__ bsent,
                     const float* __restrict__ Wrecv, const float* __restrict__ brecv,
                     const float* __restrict__ Wmsg,  const float* __restrict__ bmsg,
                     const float* __restrict__ Wself, const float* __restrict__ bself,
                     float* __restrict__ sentP, float* __restrict__ recvP,
                     float* __restrict__ msgP,  float* __restrict__ selfOut) {
    const int wave = threadIdx.x >> 5;
    const int lane = threadIdx.x & 31;
    const int lm   = lane & 15;        // N index within tile / M row for A
    const int lh   = lane >> 4;        // lane-half selector
    const int r0   = blockIdx.x * 16;  // 50000 % 16 == 0 -> always full tiles

    const float* W; const float* b; float* O;
    if      (wave == 0) { W = Wsent; b = bsent; O = sentP;  }
    else if (wave == 1) { W = Wrecv; b = brecv; O = recvP;  }
    else if (wave == 2) { W = Wmsg;  b = bmsg;  O = msgP;   }
    else                { W = Wself; b = bself; O = selfOut; }

    v8f acc0 = {}, acc1 = {}, acc2 = {}, acc3 = {};
    // A frag (16x4 fp32): lanes 0-15 rows, VGPR0/1 = K {0,1} (lh=0) or {2,3} (lh=1)
    const float* arow = nodes + (size_t)(r0 + lm) * DIM + 2 * lh;

#pragma unroll
    for (int k = 0; k < 16; ++k) {
        v2f a;
        a.x = arow[4 * k];
        a.y = arow[4 * k + 1];
        const int krow = 4 * k + 2 * lh;   // B frag mirrors A layout (4x16, K x N)
        v2f b0, b1, b2, b3;
        b0.x = W[(size_t)krow * DIM + lm];        b0.y = W[(size_t)(krow + 1) * DIM + lm];
        b1.x = W[(size_t)krow * DIM + 16 + lm];   b1.y = W[(size_t)(krow + 1) * DIM + 16 + lm];
        b2.x = W[(size_t)krow * DIM + 32 + lm];   b2.y = W[(size_t)(krow + 1) * DIM + 32 + lm];
        b3.x = W[(size_t)krow * DIM + 48 + lm];   b3.y = W[(size_t)(krow + 1) * DIM + 48 + lm];
        acc0 = __builtin_amdgcn_wmma_f32_16x16x4_f32(false, a, false, b0, (short)0, acc0, false, false);
        acc1 = __builtin_amdgcn_wmma_f32_16x16x4_f32(false, a, false, b1, (short)0, acc1, false, false);
        acc2 = __builtin_amdgcn_wmma_f32_16x16x4_f32(false, a, false, b2, (short)0, acc2, false, false);
        acc3 = __builtin_amdgcn_wmma_f32_16x16x4_f32(false, a, false, b3, (short)0, acc3, false, false);
    }

    // C/D layout: VGPR v -> row v + 8*lh, col = lm (+16*t). Add bias, store.
#pragma unroll
    for (int v = 0; v < 8; ++v) {
        const size_t row = (size_t)(r0 + v + 8 * lh) * DIM;
        O[row + lm]      = acc0[v] + b[lm];
        O[row + 16 + lm] = acc1[v] + b[16 + lm];
        O[row + 32 + lm] = acc2[v] + b[32 + lm];
        O[row + 48 + lm] = acc3[v] + b[48 + lm];
    }
}

// --------------------------------------------------------------------------
// Kernel 2: fused edge pass. Per wave: 16x64 WMMA tile of edges @ W_edge,
// + b_edge + sentP[senders] + recvP[receivers] -> edge_feat (stored to
// d_out), in-register attention logit dot (wave32 butterfly reduce),
// leaky_relu, atomicMax of encoded logit per receiver.
// --------------------------------------------------------------------------
__global__ void __launch_bounds__(256)
gat_edge_kernel(const float* __restrict__ edges,
                const int* __restrict__ senders, const int* __restrict__ receivers,
                const float* __restrict__ Wedge, const float* __restrict__ bedge,
                const float* __restrict__ Wattn, const float* __restrict__ battn,
                const float* __restrict__ sentP, const float* __restrict__ recvP,
                float* __restrict__ edgeFeat, float* __restrict__ logits,
                unsigned* __restrict__ maxkeys) {
    const int wid  = blockIdx.x * (blockDim.x >> 5) + (threadIdx.x >> 5);
    const int e0   = wid * 16;             // 800000 % 16 == 0
    const int lane = threadIdx.x & 31;
    const int lm   = lane & 15;
    const int lh   = lane >> 4;
    const int base = e0 + 8 * lh;          // first edge row handled by this half

    int se[8], re[8];
#pragma unroll
    for (int v = 0; v < 8; ++v) {
        se[v] = senders[base + v];
        re[v] = receivers[base + v];
    }

    v8f acc0 = {}, acc1 = {}, acc2 = {}, acc3 = {};
    const float* arow = edges + (size_t)(e0 + lm) * DIM + 2 * lh;
#pragma unroll
    for (int k = 0; k < 16; ++k) {
        v2f a;
        a.x = arow[4 * k];
        a.y = arow[4 * k + 1];
        const int krow = 4 * k + 2 * lh;
        v2f b0, b1, b2, b3;
        b0.x = Wedge[(size_t)krow * DIM + lm];        b0.y = Wedge[(size_t)(krow + 1) * DIM + lm];
        b1.x = Wedge[(size_t)krow * DIM + 16 + lm];   b1.y = Wedge[(size_t)(krow + 1) * DIM + 16 + lm];
        b2.x = Wedge[(size_t)krow * DIM + 32 + lm];   b2.y = Wedge[(size_t)(krow + 1) * DIM + 32 + lm];
        b3.x = Wedge[(size_t)krow * DIM + 48 + lm];   b3.y = Wedge[(size_t)(krow + 1) * DIM + 48 + lm];
        acc0 = __builtin_amdgcn_wmma_f32_16x16x4_f32(false, a, false, b0, (short)0, acc0, false, false);
        acc1 = __builtin_amdgcn_wmma_f32_16x16x4_f32(false, a, false, b1, (short)0, acc1, false, false);
        acc2 = __builtin_amdgcn_wmma_f32_16x16x4_f32(false, a, false, b2, (short)0, acc2, false, false);
        acc3 = __builtin_amdgcn_wmma_f32_16x16x4_f32(false, a, false, b3, (short)0, acc3, false, false);
    }

    // edge_feat = wmma + b_edge + sentP[s] + recvP[r]; accumulate logit partials.
    float p[8];
#pragma unroll
    for (int v = 0; v < 8; ++v) p[v] = 0.0f;

#pragma unroll
    for (int t = 0; t < 4; ++t) {
        const int col = 16 * t + lm;
        const float wa = Wattn[col];
        const float bb = bedge[col];
        const v8f acc = (t == 0) ? acc0 : (t == 1) ? acc1 : (t == 2) ? acc2 : acc3;
#pragma unroll
        for (int v = 0; v < 8; ++v) {
            float f = acc[v] + bb + sentP[(size_t)se[v] * DIM + col]
                             + recvP[(size_t)re[v] * DIM + col];
            edgeFeat[(size_t)(base + v) * DIM + col] = f;
            p[v] += f * wa;
        }
    }

    // Butterfly reduce across the 16 lanes sharing each row group.
#pragma unroll
    for (int m = 1; m < 16; m <<= 1) {
#pragma unroll
        for (int v = 0; v < 8; ++v) p[v] += __shfl_xor(p[v], m, 32);
    }

    // Lane j (<8) of each half finalizes row base+j.
    if (lm < 8) {
        float logit = p[lm] + battn[0];
        float lr = (logit > 0.0f) ? logit : 0.01f * logit;   // leaky_relu
        logits[base + lm] = lr;
        atomicMax(&maxkeys[re[lm]], enc_f32(lr));
    }
}

// --------------------------------------------------------------------------
// Kernel 3: ex = exp(logit - seg_max[recv]) in place; seg_sum via atomicAdd.
// --------------------------------------------------------------------------
__global__ void gat_softmax_kernel(const int* __restrict__ receivers,
                                   const unsigned* __restrict__ maxkeys,
                                   float* __restrict__ logits,
                                   float* __restrict__ sums) {
    int e = blockIdx.x * blockDim.x + threadIdx.x;
    if (e >= N_EDGES) return;
    int r = receivers[e];
    float m  = dec_f32(maxkeys[r]);
    float ex = __expf(logits[e] - m);
    logits[e] = ex;
    atomicAdd(&sums[r], ex);
}

// --------------------------------------------------------------------------
// Kernel 4: new_nodes[r] += (ex/sum[r]) * msgP[s]; one lane per (edge,col).
// d_out already holds the self-projection seed from kernel 1.
// --------------------------------------------------------------------------
__global__ void gat_aggregate_kernel(const int* __restrict__ senders,
                                     const int* __restrict__ receivers,
                                     const float* __restrict__ logits,
                                     const float* __restrict__ sums,
                                     const float* __restrict__ msgP,
                                     float* __restrict__ newNodes) {
    int t = blockIdx.x * blockDim.x + threadIdx.x;   // < E*64 = 51.2M
    int e = t >> 6;
    int c = t & 63;
    int r = receivers[e];
    int s = senders[e];
    float attn = logits[e] / sums[r];
    atomicAdd(&newNodes[(size_t)r * DIM + c], attn * msgP[(size_t)s * DIM + c]);
}

// --------------------------------------------------------------------------
extern "C" void kernel_launch(void* const* d_in, const int* in_sizes, int n_in,
                              void* d_out, int out_size, void* d_ws, size_t ws_size,
                              hipStream_t stream) {
    const float* nodes     = (const float*)d_in[0];
    const float* edges     = (const float*)d_in[1];
    const int*   senders   = (const int*)d_in[2];
    const int*   receivers = (const int*)d_in[3];
    const float* Wsent = (const float*)d_in[4];  const float* bsent = (const float*)d_in[5];
    const float* Wrecv = (const float*)d_in[6];  const float* brecv = (const float*)d_in[7];
    const float* Wedge = (const float*)d_in[8];  const float* bedge = (const float*)d_in[9];
    const float* Wattn = (const float*)d_in[10]; const float* battn = (const float*)d_in[11];
    const float* Wmsg  = (const float*)d_in[12]; const float* bmsg  = (const float*)d_in[13];
    const float* Wself = (const float*)d_in[14]; const float* bself = (const float*)d_in[15];

    const size_t NP = (size_t)N_NODES * DIM;     // 3.2M floats
    float* newNodes = (float*)d_out;             // [N_NODES, 64]
    float* edgeFeat = newNodes + NP;             // [N_EDGES, 64]

    // Workspace layout (~42 MB): sentP | recvP | msgP | logits | maxkeys | sums
    float*    ws      = (float*)d_ws;
    float*    sentP   = ws;
    float*    recvP   = ws + NP;
    float*    msgP    = ws + 2 * NP;
    float*    logits  = ws + 3 * NP;
    unsigned* maxkeys = (unsigned*)(ws + 3 * NP + N_EDGES);
    float*    sums    = (float*)(maxkeys + N_NODES);

    // 0) init softmax state
    gat_init_kernel<<<(N_NODES + 255) / 256, 256, 0, stream>>>(maxkeys, sums);

    // 1) node projections (4 GEMMs); selfP seeds d_out
    gat_node_proj_kernel<<<N_NODES / 16, 128, 0, stream>>>(
        nodes, Wsent, bsent, Wrecv, brecv, Wmsg, bmsg, Wself, bself,
        sentP, recvP, msgP, newNodes);

    // 2) fused edge pass: GEMM + gathers + edge_feat + logits + segment max
    gat_edge_kernel<<<N_EDGES / 128, 256, 0, stream>>>(
        edges, senders, receivers, Wedge, bedge, Wattn, battn,
        sentP, recvP, edgeFeat, logits, maxkeys);

    // 3) segment softmax: exp & sums
    gat_softmax_kernel<<<(N_EDGES + 255) / 256, 256, 0, stream>>>(
        receivers, maxkeys, logits, sums);

    // 4) weighted message scatter-add
    gat_aggregate_kernel<<<(N_EDGES * 64) / 256, 256, 0, stream>>>(
        senders, receivers, logits, sums, msgP, newNodes);
}